// SQSubM2d_34651796144690
// MI455X (gfx1250) — compile-verified
//
#include <hip/hip_runtime.h>
#include <math.h>

#define HH 512
#define WW 512
#define CC 64
#define COUT 64
#define KSIZE 576          // C*3*3
#define QMAX 127.0f
#define TSTRIDE 65         // padded LDS row stride (dwords) -> conflict-free transpose

// ---------------- workspace layout (floats) ----------------
// [0..575]     act_scale (atomic-max area, zero-initialized each call)
// [576..1151]  scale
// [1152..1727] Mx = 1/(scale * s_x)
// [1728]       s_x

__global__ void k0_init(float* __restrict__ ws) {
    int t = threadIdx.x;
    if (t < KSIZE) ws[t] = 0.0f;
}

// One block per (channel, 32-row chunk): 9 masked window maxes -> atomicMax.
__global__ void k1_act_scale(const float* __restrict__ in, float* __restrict__ ws_act) {
    __shared__ float red[9][256];
    int bid = blockIdx.x;
    int c  = bid >> 4;
    int r0 = (bid & 15) << 5;          // 32 rows per chunk
    int t  = threadIdx.x;

    float m[9];
#pragma unroll
    for (int i = 0; i < 9; ++i) m[i] = 0.0f;

    const float* base = in + (size_t)c * (HH * WW);
    for (int idx = t; idx < 32 * WW; idx += 256) {
        int h = r0 + (idx >> 9);
        int w = idx & 511;
        float a = fabsf(base[h * WW + w]);
        bool r0ok = (h <= HH - 2), r2ok = (h >= 1);   // kh=0 / kh=2 row windows
        bool c0ok = (w <= WW - 2), c2ok = (w >= 1);   // kw=0 / kw=2 col windows
        float a0 = r0ok ? a : 0.0f;
        float a2 = r2ok ? a : 0.0f;
        m[0] = fmaxf(m[0], c0ok ? a0 : 0.0f);
        m[1] = fmaxf(m[1], a0);
        m[2] = fmaxf(m[2], c2ok ? a0 : 0.0f);
        m[3] = fmaxf(m[3], c0ok ? a  : 0.0f);
        m[4] = fmaxf(m[4], a);
        m[5] = fmaxf(m[5], c2ok ? a  : 0.0f);
        m[6] = fmaxf(m[6], c0ok ? a2 : 0.0f);
        m[7] = fmaxf(m[7], a2);
        m[8] = fmaxf(m[8], c2ok ? a2 : 0.0f);
    }
#pragma unroll
    for (int i = 0; i < 9; ++i) red[i][t] = m[i];
    __syncthreads();
    for (int s = 128; s > 0; s >>= 1) {
        if (t < s) {
#pragma unroll
            for (int i = 0; i < 9; ++i)
                red[i][t] = fmaxf(red[i][t], red[i][t + s]);
        }
        __syncthreads();
    }
    if (t < 9) {
        // values are >= 0 and slot init is 0.0f, so uint compare == float compare
        atomicMax((unsigned int*)&ws_act[c * 9 + t], __float_as_uint(red[t][0]));
    }
}

// Single block: w_scale, smooth scale, amax_x/amax_w, s_x/Mx to ws, wq output.
__global__ void k2_scales_wq(const float* __restrict__ wgt, float* __restrict__ ws,
                             float* __restrict__ wq_out) {
    __shared__ float red[1024];
    __shared__ float sscale[KSIZE];
    int t = threadIdx.x;

    float act = 0.0f, wmax = 0.0f, s = 1.0f;
    if (t < KSIZE) {
        act = ws[t];
#pragma unroll 4
        for (int co = 0; co < COUT; ++co)
            wmax = fmaxf(wmax, fabsf(wgt[co * KSIZE + t]));
        s = sqrtf(act) / sqrtf(wmax);     // alpha = 0.5
        if (s == 0.0f) s = 1.0f;
        sscale[t] = s;
    }

    red[t] = (t < KSIZE) ? act / s : 0.0f;          // amax of scaled activations
    __syncthreads();
    for (int st = 512; st > 0; st >>= 1) {
        if (t < st) red[t] = fmaxf(red[t], red[t + st]);
        __syncthreads();
    }
    float amax_x = red[0];
    __syncthreads();
    red[t] = (t < KSIZE) ? wmax * s : 0.0f;         // amax of scaled weights
    __syncthreads();
    for (int st = 512; st > 0; st >>= 1) {
        if (t < st) red[t] = fmaxf(red[t], red[t + st]);
        __syncthreads();
    }
    float amax_w = red[0];
    __syncthreads();

    float s_x = (amax_x > 0.0f) ? amax_x * (1.0f / QMAX) : 1.0f;
    float s_w = (amax_w > 0.0f) ? amax_w * (1.0f / QMAX) : 1.0f;
    float inv_sx = 1.0f / s_x;
    float inv_sw = 1.0f / s_w;
    if (t == 0) ws[1728] = s_x;
    if (t < KSIZE) {
        ws[576 + t]  = s;
        ws[1152 + t] = (1.0f / s) * inv_sx;         // combined multiplier Mx
    }

    // wq[cout][kh][kw][c] = FQ(weight[cout][c][kh][kw] * scale[j]; s_w)
    for (int o = t; o < COUT * KSIZE; o += 1024) {
        int cout = o / KSIZE;
        int idx  = o - cout * KSIZE;                // (kh*3+kw)*64 + c
        int kh = idx / 192;
        int kw = (idx / 64) % 3;
        int cj = idx & 63;
        int j  = cj * 9 + kh * 3 + kw;
        float v = wgt[(cout * CC + cj) * 9 + kh * 3 + kw] * sscale[j];
        float q = rintf(fminf(fmaxf(v * inv_sw, -QMAX), QMAX)) * s_w;
        wq_out[o] = q;
    }
}

// Main streaming pass: async global->LDS staging + transpose + 9-term fold.
__global__ void k3_fold(const float* __restrict__ in, const float* __restrict__ ws,
                        float* __restrict__ xf) {
    __shared__ float tile[CC * TSTRIDE];
    int bid = blockIdx.x;
    int h  = bid >> 3;
    int w0 = (bid & 7) << 6;
    int t  = threadIdx.x;

    // ---- async copy: 64 channels x 64 w of row h into LDS (ASYNCcnt path) ----
    const float* src = in + (size_t)h * WW + w0;
#pragma unroll
    for (int it = 0; it < 16; ++it) {
        int q = t + (it << 8);             // 0..4095
        int c = q >> 6;
        int w = q & 63;
        const float* gp = src + (size_t)c * (HH * WW) + w;
        float* lp = &tile[c * TSTRIDE + w];
#if __has_builtin(__builtin_amdgcn_global_load_async_to_lds_b32)
        __builtin_amdgcn_global_load_async_to_lds_b32(
            (__attribute__((address_space(1))) int*)(void*)gp,
            (__attribute__((address_space(3))) int*)(void*)lp, 0, 0);
#else
        unsigned lds_off = (unsigned)(unsigned long long)
            (__attribute__((address_space(3))) float*)lp;
        unsigned long long ga = (unsigned long long)gp;
        asm volatile("global_load_async_to_lds_b32 %0, %1, off"
                     :: "v"(lds_off), "v"(ga) : "memory");
#endif
    }
#if __has_builtin(__builtin_amdgcn_s_wait_asynccnt)
    __builtin_amdgcn_s_wait_asynccnt(0);
#else
    asm volatile("s_wait_asynccnt 0x0" ::: "memory");
#endif
    __syncthreads();

    // ---- compute: each thread owns 4 fixed channels, 4 w positions ----
    int cbase = (t & 15) << 2;
    int wl    = t >> 4;
    float mx[36];
    const float* wsM = ws + 1152;
#pragma unroll
    for (int i = 0; i < 36; ++i) mx[i] = wsM[cbase * 9 + i];
    float s_x = ws[1728];

    bool r0ok = (h <= HH - 2), r2ok = (h >= 1);

    for (int it = 0; it < 4; ++it) {
        int w  = wl + (it << 4);
        int wg = w0 + w;
        bool c0ok = (wg <= WW - 2), c2ok = (wg >= 1);
        float4 outv;
        float* po = (float*)&outv;
#pragma unroll
        for (int ci = 0; ci < 4; ++ci) {
            float v = tile[(cbase + ci) * TSTRIDE + w];
            float acc = 0.0f;
#pragma unroll
            for (int kh = 0; kh < 3; ++kh) {
                bool rok = (kh == 0) ? r0ok : ((kh == 2) ? r2ok : true);
#pragma unroll
                for (int kw = 0; kw < 3; ++kw) {
                    bool cok = (kw == 0) ? c0ok : ((kw == 2) ? c2ok : true);
                    float p = v * mx[ci * 9 + kh * 3 + kw];
                    float r = rintf(fminf(fmaxf(p, -QMAX), QMAX));
                    acc += (rok && cok) ? r : 0.0f;
                }
            }
            po[ci] = acc * s_x;
        }
        *(float4*)(xf + ((size_t)(h * WW + wg) * CC + cbase)) = outv;
    }
}

extern "C" void kernel_launch(void* const* d_in, const int* in_sizes, int n_in,
                              void* d_out, int out_size, void* d_ws, size_t ws_size,
                              hipStream_t stream) {
    const float* inp = (const float*)d_in[0];   // (1,64,512,512) f32
    const float* wgt = (const float*)d_in[1];   // (64,64,3,3) f32
    float* out = (float*)d_out;
    float* ws  = (float*)d_ws;
    float* wq  = out;                           // 36864 floats
    float* xf  = out + COUT * 3 * 3 * CC;       // 16,777,216 floats (NHWC)

    hipLaunchKernelGGL(k0_init,      dim3(1),    dim3(576),  0, stream, ws);
    hipLaunchKernelGGL(k1_act_scale, dim3(1024), dim3(256),  0, stream, inp, ws);
    hipLaunchKernelGGL(k2_scales_wq, dim3(1),    dim3(1024), 0, stream, wgt, ws, wq);
    hipLaunchKernelGGL(k3_fold,      dim3(4096), dim3(256),  0, stream, inp, ws, xf);
}